// Resampling_11974368821422
// MI455X (gfx1250) — compile-verified
//
#include <hip/hip_runtime.h>

typedef __attribute__((ext_vector_type(2))) float v2f;
typedef __attribute__((ext_vector_type(4))) float v4f;
typedef __attribute__((ext_vector_type(8))) float v8f;

#define HH   32
#define WW   32
#define DD   32
#define CC   8
#define HWD  32768          // 32*32*32
#define CHUNKS 8            // blocks per (b,p) pair
#define PTS_PER_BLOCK 4096  // HWD / CHUNKS

__global__ __launch_bounds__(256) void resample3d_wmma_kernel(
    const float* __restrict__ fmap,   // (B,P,H,W,D,C) f32
    const float* __restrict__ theta,  // (B, 12*P)     f32
    float* __restrict__ out)          // (B,P,H,W,D,C) f32
{
    const int lane = threadIdx.x & 31;
    const int wave = threadIdx.x >> 5;
    const int bp    = blockIdx.x / CHUNKS;   // b*P + p
    const int chunk = blockIdx.x % CHUNKS;

    // ---------- A matrix: theta(3x4) zero-padded to 16x4 ----------
    // f32 16x4 A layout: lanes 0-15 hold (M=lane, K=0..1) in v[0..1],
    //                    lanes 16-31 hold (M=lane-16, K=2..3).
    const float* th = theta + bp * 12;       // (b*96 + p*12)
    const int m  = lane & 15;
    const int kb = (lane < 16) ? 0 : 2;
    v2f amat;
    amat[0] = (m < 3) ? th[m * 4 + kb + 0] : 0.0f;
    amat[1] = (m < 3) ? th[m * 4 + kb + 1] : 0.0f;

    const v4f* fbase = (const v4f*)fmap + (size_t)bp * (HWD * 2);
    v4f*       obase = (v4f*)out        + (size_t)bp * (HWD * 2);

    const int base0 = chunk * PTS_PER_BLOCK + wave * 32;
    const float inv31x2 = 2.0f / 31.0f;

    for (int t = 0; t < 16; ++t) {
        const int pbase = base0 + t * 256;

        // ---------- B matrices: grid columns (x,y,z,1) for 2x16 points ----
        // f32 4x16 B layout: v[0]: lanes 0-15 = row K=0 (x), lanes 16-31 = K=2 (y)
        //                    v[1]: lanes 0-15 = row K=1 (z), lanes 16-31 = K=3 (1)
        v2f b0, b1;
        {
            int n  = pbase + (lane & 15);
            int dz = n & 31, wx = (n >> 5) & 31, hy = (n >> 10) & 31;
            float xg = fmaf((float)wx, inv31x2, -1.0f);
            float yg = fmaf((float)hy, inv31x2, -1.0f);
            float zg = fmaf((float)dz, inv31x2, -1.0f);
            b0[0] = (lane < 16) ? xg : yg;
            b0[1] = (lane < 16) ? zg : 1.0f;
        }
        {
            int n  = pbase + 16 + (lane & 15);
            int dz = n & 31, wx = (n >> 5) & 31, hy = (n >> 10) & 31;
            float xg = fmaf((float)wx, inv31x2, -1.0f);
            float yg = fmaf((float)hy, inv31x2, -1.0f);
            float zg = fmaf((float)dz, inv31x2, -1.0f);
            b1[0] = (lane < 16) ? xg : yg;
            b1[1] = (lane < 16) ? zg : 1.0f;
        }

        // ---------- coords = theta x grid via V_WMMA_F32_16X16X4_F32 ------
        v8f c0 = {};
        v8f c1 = {};
        c0 = __builtin_amdgcn_wmma_f32_16x16x4_f32(
                 false, amat, false, b0, (short)0, c0, false, false);
        c1 = __builtin_amdgcn_wmma_f32_16x16x4_f32(
                 false, amat, false, b1, (short)0, c1, false, false);

        // D rows 0..2 (x,y,z) live in VGPR 0..2 of lanes 0..15.
        // Lanes 16..31 pull their point's coords out of the second WMMA
        // result via a wave32 shuffle from lane (lane-16).
        float xB = __shfl(c1[0], lane & 15, 32);
        float yB = __shfl(c1[1], lane & 15, 32);
        float zB = __shfl(c1[2], lane & 15, 32);
        bool lo = (lane < 16);
        float xv = lo ? c0[0] : xB;
        float yv = lo ? c0[1] : yB;
        float zv = lo ? c0[2] : zB;

        // ---------- trilinear sample (reference semantics) ---------------
        float x = 0.5f * (xv + 1.0f) * (float)(WW - 2);
        float y = 0.5f * (yv + 1.0f) * (float)(HH - 2);
        float z = 0.5f * (zv + 1.0f) * (float)(DD - 2);

        // clamp in float first so the int conversion is well-defined
        float xf = floorf(x), yf = floorf(y), zf = floorf(z);
        int x0 = min(max((int)fminf(fmaxf(xf, -1.0f), 32.0f), 0), WW - 1);
        int x1 = min(max((int)fminf(fmaxf(xf + 1.0f, -1.0f), 32.0f), 0), WW - 1);
        int y0 = min(max((int)fminf(fmaxf(yf, -1.0f), 32.0f), 0), HH - 1);
        int y1 = min(max((int)fminf(fmaxf(yf + 1.0f, -1.0f), 32.0f), 0), HH - 1);
        int z0 = min(max((int)fminf(fmaxf(zf, -1.0f), 32.0f), 0), DD - 1);
        int z1 = min(max((int)fminf(fmaxf(zf + 1.0f, -1.0f), 32.0f), 0), DD - 1);

        float xd = x - (float)x0;               // uses clipped x0 (matches ref)
        float yd = (float)y1 - y;               // ref: yd = y1 - y
        float zd = z - (float)z0;
        float xm = 1.0f - xd, ym = 1.0f - yd, zm = 1.0f - zd;

        v4f r0 = {0.0f, 0.0f, 0.0f, 0.0f};
        v4f r1 = {0.0f, 0.0f, 0.0f, 0.0f};

        // corner gather: each voxel = 2 x b128 (C = 8 floats); these re-read
        // the fmap slice ~8x, so they should stay resident in the 192MB L2.
        #define CORNER(yy, xx, zz, wgt)                                   \
        {                                                                 \
            size_t v = ((size_t)(((yy) * WW + (xx)) * DD + (zz))) * 2;    \
            v4f p0 = fbase[v];                                            \
            v4f p1 = fbase[v + 1];                                        \
            float wv = (wgt);                                             \
            r0 += p0 * wv;                                                \
            r1 += p1 * wv;                                                \
        }
        CORNER(y1, x0, z0, xm * ym * zm);   // c000
        CORNER(y1, x1, z0, xd * ym * zm);   // c100
        CORNER(y0, x0, z0, xm * yd * zm);   // c010
        CORNER(y1, x0, z1, xm * ym * zd);   // c001
        CORNER(y1, x1, z1, xd * ym * zd);   // c101
        CORNER(y0, x0, z1, xm * yd * zd);   // c011
        CORNER(y0, x1, z0, xd * yd * zm);   // c110
        CORNER(y0, x1, z1, xd * yd * zd);   // c111
        #undef CORNER

        // Output is written once and never re-read: stream it past L2 with
        // non-temporal stores (TH=NT) so it does not evict the fmap slices
        // that the gathers are hitting in L2.
        const int np = pbase + lane;
        __builtin_nontemporal_store(r0, &obase[(size_t)np * 2 + 0]);
        __builtin_nontemporal_store(r1, &obase[(size_t)np * 2 + 1]);
    }
}

extern "C" void kernel_launch(void* const* d_in, const int* in_sizes, int n_in,
                              void* d_out, int out_size, void* d_ws, size_t ws_size,
                              hipStream_t stream) {
    (void)in_sizes; (void)n_in; (void)out_size; (void)d_ws; (void)ws_size;
    const float* fmap  = (const float*)d_in[0];
    const float* theta = (const float*)d_in[1];
    float* out = (float*)d_out;

    const int B = 16, P = 8;
    dim3 grid(B * P * CHUNKS);   // 1024 blocks
    dim3 block(256);             // 8 wave32 waves
    resample3d_wmma_kernel<<<grid, block, 0, stream>>>(fmap, theta, out);
}